// GATConv_30640296690058
// MI455X (gfx1250) — compile-verified
//
#include <hip/hip_runtime.h>
#include <hip/hip_bf16.h>

typedef __attribute__((ext_vector_type(2))) float v2f;
typedef __attribute__((ext_vector_type(8))) float v8f;

#define N_NODES   8192
#define IN_FEAT   512
#define OUT_FEAT  64
#define NEG_SLOPE 0.01f

__device__ __forceinline__ float lrelu(float x) { return x > 0.0f ? x : NEG_SLOPE * x; }

// order-preserving float <-> uint map so we can use integer atomicMax
__device__ __forceinline__ unsigned f2ord(float f) {
    unsigned u = __float_as_uint(f);
    return (u & 0x80000000u) ? ~u : (u | 0x80000000u);
}
__device__ __forceinline__ float ord2f(unsigned u) {
    unsigned b = (u & 0x80000000u) ? (u & 0x7fffffffu) : ~u;
    return __uint_as_float(b);
}

// ---------------------------------------------------------------------------
// Kernel 1: h = X (8192x512) @ W (512x64), fp32 WMMA 16x16x4.
// Grid: 512 blocks of 128 threads. Wave w computes the 16x16 tile at
// rows [blk*16, +16), cols [w*16, +16).
// A layout (ISA 32-bit A 16x4): lanes 0-15 hold M=lane, K={k,k+1};
// lanes 16-31 hold M=lane-16, K={k+2,k+3}.  B symmetric.  C: VGPR v ->
// M = v + 8*(lane>>4), N = lane&15.
// ---------------------------------------------------------------------------
__global__ void __launch_bounds__(128) gat_gemm_h(const float* __restrict__ X,
                                                  const float* __restrict__ W,
                                                  float* __restrict__ h) {
    const int tid  = threadIdx.x;
    const int wave = tid >> 5;
    const int lane = tid & 31;
    const int half = lane >> 4;
    const int r    = lane & 15;
    const int row0 = blockIdx.x * 16;
    const int n0   = wave * 16;

    const float* xrow = X + (size_t)(row0 + r) * IN_FEAT;
    v8f c = {};
    for (int k = 0; k < IN_FEAT; k += 4) {
        const int ka = k + 2 * half;
        v2f a;
        a.x = xrow[ka];
        a.y = xrow[ka + 1];
        v2f b;
        b.x = W[(size_t)ka * OUT_FEAT + n0 + r];
        b.y = W[(size_t)(ka + 1) * OUT_FEAT + n0 + r];
        c = __builtin_amdgcn_wmma_f32_16x16x4_f32(false, a, false, b, (short)0, c, false, false);
    }
#pragma unroll
    for (int v = 0; v < 8; ++v)
        h[(size_t)(row0 + v + 8 * half) * OUT_FEAT + n0 + r] = c[v];
}

// ---------------------------------------------------------------------------
// Kernel 2 helpers: init the global-max cell, then per-row attention scores.
// ---------------------------------------------------------------------------
__global__ void gat_init(unsigned* maxu) { *maxu = 0u; }

__global__ void gat_score(const float* __restrict__ h, const float* __restrict__ att,
                          float* __restrict__ s_src, float* __restrict__ s_dst,
                          unsigned* __restrict__ maxu) {
    const int i = blockIdx.x * blockDim.x + threadIdx.x;
    if (i >= N_NODES) return;
    const float* hr = h + (size_t)i * OUT_FEAT;
    float a = 0.f, b = 0.f;
#pragma unroll 8
    for (int f = 0; f < OUT_FEAT; ++f) {
        const float hv = hr[f];
        a = fmaf(hv, att[f], a);
        b = fmaf(hv, att[OUT_FEAT + f], b);
    }
    s_src[i] = a;
    s_dst[i] = b;
    atomicMax(maxu, f2ord(b));
}

// ---------------------------------------------------------------------------
// Kernel 3: streaming softmax-weighted aggregation (flash-attention style),
// 32 output rows per block so each set of B loads feeds 8 WMMAs.
// h'(i,:) = sum_j exp(lrelu(s_src_i + s_dst_j) - m_i) * h(j,:) / denom_i
// with m_i = lrelu(s_src_i + max_j s_dst_j)  (exact row max by monotonicity).
// Grid: 256 blocks (32 rows each) x 128 threads. Waves split K=8192 into
// quarters; each wave holds 8 C tiles (two 16-row strips x four 16-col
// tiles) + two per-lane denominators. Cross-wave reduce via ds_add_f32.
// ---------------------------------------------------------------------------
#define PF_H   4096   // prefetch h 4 K-steps (16 rows) ahead, in bytes
#define PF_S   1024   // prefetch s_dst 256 elements ahead, in bytes

__global__ void __launch_bounds__(128) gat_attn(const float* __restrict__ h,
                                                const float* __restrict__ s_src,
                                                const float* __restrict__ s_dst,
                                                const unsigned* __restrict__ maxu,
                                                float* __restrict__ out) {
    __shared__ float lds_acc[32 * OUT_FEAT];
    __shared__ float lds_den[32];

    const int tid  = threadIdx.x;
    const int wave = tid >> 5;
    const int lane = tid & 31;
    const int half = lane >> 4;
    const int r    = lane & 15;
    const int row0 = blockIdx.x * 32;

    const float mx    = ord2f(*maxu);
    const float ssrc0 = s_src[row0 + r];
    const float ssrc1 = s_src[row0 + 16 + r];
    const float m0    = lrelu(ssrc0 + mx);
    const float m1    = lrelu(ssrc1 + mx);

    v8f c00 = {}, c01 = {}, c02 = {}, c03 = {};   // rows [row0, row0+16)
    v8f c10 = {}, c11 = {}, c12 = {}, c13 = {};   // rows [row0+16, row0+32)
    float den0 = 0.f, den1 = 0.f;

    const int kbeg = wave * (N_NODES / 4);
    const int kend = kbeg + (N_NODES / 4);
    for (int k = kbeg; k < kend; k += 4) {
        const int ka = k + 2 * half;           // this lane's K pair {ka, ka+1}
        const float d0 = s_dst[ka];
        const float d1 = s_dst[ka + 1];

        const float p00 = __expf(lrelu(ssrc0 + d0) - m0);
        const float p01 = __expf(lrelu(ssrc0 + d1) - m0);
        const float p10 = __expf(lrelu(ssrc1 + d0) - m1);
        const float p11 = __expf(lrelu(ssrc1 + d1) - m1);
        den0 += p00 + p01;
        den1 += p10 + p11;
        v2f a0; a0.x = p00; a0.y = p01;        // A = P tile, row strip 0
        v2f a1; a1.x = p10; a1.y = p11;        // A = P tile, row strip 1

        const float* h0 = h + (size_t)ka * OUT_FEAT + r;
        const float* h1 = h0 + OUT_FEAT;
        v2f b0; b0.x = h0[0];  b0.y = h1[0];   // B = H tile (4 j x 16 feats)
        v2f b1; b1.x = h0[16]; b1.y = h1[16];
        v2f b2; b2.x = h0[32]; b2.y = h1[32];
        v2f b3; b3.x = h0[48]; b3.y = h1[48];

        // keep L0/WGP$ ahead of the WMMA pipeline (global_prefetch_b8)
        __builtin_prefetch((const char*)h0 + PF_H, 0, 3);
        __builtin_prefetch((const char*)(s_dst + ka) + PF_S, 0, 3);

        c00 = __builtin_amdgcn_wmma_f32_16x16x4_f32(false, a0, false, b0, (short)0, c00, false, false);
        c01 = __builtin_amdgcn_wmma_f32_16x16x4_f32(false, a0, false, b1, (short)0, c01, false, false);
        c02 = __builtin_amdgcn_wmma_f32_16x16x4_f32(false, a0, false, b2, (short)0, c02, false, false);
        c03 = __builtin_amdgcn_wmma_f32_16x16x4_f32(false, a0, false, b3, (short)0, c03, false, false);
        c10 = __builtin_amdgcn_wmma_f32_16x16x4_f32(false, a1, false, b0, (short)0, c10, false, false);
        c11 = __builtin_amdgcn_wmma_f32_16x16x4_f32(false, a1, false, b1, (short)0, c11, false, false);
        c12 = __builtin_amdgcn_wmma_f32_16x16x4_f32(false, a1, false, b2, (short)0, c12, false, false);
        c13 = __builtin_amdgcn_wmma_f32_16x16x4_f32(false, a1, false, b3, (short)0, c13, false, false);
    }

    for (int i = tid; i < 32 * OUT_FEAT; i += 128) lds_acc[i] = 0.f;
    if (tid < 32) lds_den[tid] = 0.f;
    __syncthreads();

    atomicAdd(&lds_den[r], den0);
    atomicAdd(&lds_den[16 + r], den1);
#pragma unroll
    for (int v = 0; v < 8; ++v) {
        const int mr0 = v + 8 * half;          // row strip 0: LDS rows 0..15
        const int mr1 = 16 + mr0;              // row strip 1: LDS rows 16..31
        atomicAdd(&lds_acc[mr0 * OUT_FEAT + 0  + r], c00[v]);
        atomicAdd(&lds_acc[mr0 * OUT_FEAT + 16 + r], c01[v]);
        atomicAdd(&lds_acc[mr0 * OUT_FEAT + 32 + r], c02[v]);
        atomicAdd(&lds_acc[mr0 * OUT_FEAT + 48 + r], c03[v]);
        atomicAdd(&lds_acc[mr1 * OUT_FEAT + 0  + r], c10[v]);
        atomicAdd(&lds_acc[mr1 * OUT_FEAT + 16 + r], c11[v]);
        atomicAdd(&lds_acc[mr1 * OUT_FEAT + 32 + r], c12[v]);
        atomicAdd(&lds_acc[mr1 * OUT_FEAT + 48 + r], c13[v]);
    }
    __syncthreads();

    for (int i = tid; i < 32 * OUT_FEAT; i += 128) {
        const int mrow = i >> 6;
        const float v = lds_acc[i] / lds_den[mrow];
        out[(size_t)(row0 + mrow) * OUT_FEAT + (i & 63)] = (v > 0.f) ? v : expm1f(v);
    }
}

// ---------------------------------------------------------------------------
extern "C" void kernel_launch(void* const* d_in, const int* in_sizes, int n_in,
                              void* d_out, int out_size, void* d_ws, size_t ws_size,
                              hipStream_t stream) {
    (void)in_sizes; (void)n_in; (void)out_size; (void)ws_size;
    const float* X   = (const float*)d_in[0];
    // d_in[1] = adj: unused by the reference computation (dense attention).
    const float* W   = (const float*)d_in[2];
    const float* att = (const float*)d_in[3];
    float* out = (float*)d_out;

    float*    h     = (float*)d_ws;                       // 8192*64 f32 (2 MB)
    float*    s_src = h + (size_t)N_NODES * OUT_FEAT;     // 8192 f32
    float*    s_dst = s_src + N_NODES;                    // 8192 f32
    unsigned* mxu   = (unsigned*)(s_dst + N_NODES);       // 1 u32

    gat_init<<<1, 1, 0, stream>>>(mxu);
    gat_gemm_h<<<N_NODES / 16, 128, 0, stream>>>(X, W, h);
    gat_score<<<N_NODES / 256, 256, 0, stream>>>(h, att, s_src, s_dst, mxu);
    gat_attn<<<N_NODES / 32, 128, 0, stream>>>(h, s_src, s_dst, mxu, out);
}